// GraphNodePredictor_43121471652154
// MI455X (gfx1250) — compile-verified
//
#include <hip/hip_runtime.h>
#include <hip/hip_bf16.h>

typedef __attribute__((ext_vector_type(2))) float v2f;
typedef __attribute__((ext_vector_type(8))) float v8f;

#define N_NODES 100000
#define N_EDGES 1000000
#define HIDDEN 64
#define NUM_CLASSES 100
#define NUM_GRAPHS 32

// ---------------- degree / norm ----------------
__global__ void init_deg(float* __restrict__ deg, int n) {
    int i = blockIdx.x * blockDim.x + threadIdx.x;
    if (i < n) deg[i] = 1.0f;                 // self-loop contributes 1
}
__global__ void zero_counts(int* __restrict__ counts) {
    if (threadIdx.x < NUM_GRAPHS) counts[threadIdx.x] = 0;
}
__global__ void edge_deg(const int* __restrict__ dst, float* __restrict__ deg, int nE) {
    int e = blockIdx.x * blockDim.x + threadIdx.x;
    if (e < nE) atomicAdd(&deg[dst[e]], 1.0f);
}
__global__ void make_dinv(float* __restrict__ deg, int n) {
    int i = blockIdx.x * blockDim.x + threadIdx.x;
    if (i < n) { float d = deg[i]; deg[i] = (d > 0.0f) ? rsqrtf(d) : 0.0f; }
}

// ---------------- batch offset / gather index ----------------
__global__ void batch_hist(const int* __restrict__ batch, int* __restrict__ counts, int n) {
    int i = blockIdx.x * blockDim.x + threadIdx.x;
    if (i < n) atomicAdd(&counts[batch[i]], 1);
}
__global__ void compute_gidx(const int* __restrict__ counts, const int* __restrict__ batch,
                             const int* __restrict__ mni, int* __restrict__ gidx, int n) {
    int g = threadIdx.x;
    if (g >= NUM_GRAPHS) return;
    int m = mni[g];
    int bi = batch[m];
    int off = 0;
    for (int j = 0; j < bi; j++) off += counts[j];   // exclusive prefix = off_per_graph[bi]
    int gi = m + off;
    gi = gi < 0 ? 0 : (gi > n - 1 ? n - 1 : gi);
    gidx[g] = gi;
}

// ---------------- dense GEMM: g = dinv .* (A @ W), accum = g ----------------
// One wave -> 16x64 tile via 4x v_wmma_f32_16x16x4_f32 accumulators, K stepped by 4.
// A-frag (16x4 f32): lane L holds A[L&15][2*(L>>4) + {0,1}]
// B-frag (4x16 f32): lane L holds W[2*(L>>4) + {0,1}][n16 + (L&15)]
// C/D (16x16 f32): VGPR r, lanes0-15 -> M=r, lanes16-31 -> M=r+8
template <int K>
__global__ __launch_bounds__(256) void gcn_gemm(const float* __restrict__ A,
                                                const float* __restrict__ W,
                                                const float* __restrict__ dinv,
                                                float* __restrict__ g,
                                                float* __restrict__ accum, int n) {
    __shared__ float Ws[K * 64];
    for (int i = threadIdx.x; i < K * 64; i += 256) Ws[i] = W[i];
    __syncthreads();

    int tile = blockIdx.x * 8 + (threadIdx.x >> 5);   // 16-row tile per wave
    if (tile * 16 >= n) return;                       // wave-uniform: EXEC stays all-1s
    int lane = threadIdx.x & 31;
    int half = lane >> 4;
    int l16  = lane & 15;
    int row0 = tile * 16;

    const float* arow = A + (size_t)(row0 + l16) * K;
    v8f c0 = {}, c1 = {}, c2 = {}, c3 = {};

    for (int kk = 0; kk < K; kk += 4) {
        int ka = kk + 2 * half;
        v2f a;  a.x = arow[ka];          a.y = arow[ka + 1];
        const float* w0 = Ws + (size_t)ka * 64;
        const float* w1 = Ws + (size_t)(ka + 1) * 64;
        v2f b0; b0.x = w0[l16 +  0];     b0.y = w1[l16 +  0];
        v2f b1; b1.x = w0[l16 + 16];     b1.y = w1[l16 + 16];
        v2f b2; b2.x = w0[l16 + 32];     b2.y = w1[l16 + 32];
        v2f b3; b3.x = w0[l16 + 48];     b3.y = w1[l16 + 48];
        c0 = __builtin_amdgcn_wmma_f32_16x16x4_f32(false, a, false, b0, (short)0, c0, false, false);
        c1 = __builtin_amdgcn_wmma_f32_16x16x4_f32(false, a, false, b1, (short)0, c1, false, false);
        c2 = __builtin_amdgcn_wmma_f32_16x16x4_f32(false, a, false, b2, (short)0, c2, false, false);
        c3 = __builtin_amdgcn_wmma_f32_16x16x4_f32(false, a, false, b3, (short)0, c3, false, false);
    }

    // Epilogue: scale each row by dinv (source-side norm), seed accum with the
    // self-loop contribution (accum = g).
    #pragma unroll
    for (int r = 0; r < 8; r++) {
        int m = row0 + r + 8 * half;
        float s = dinv[m];
        float v0 = c0[r] * s, v1 = c1[r] * s, v2 = c2[r] * s, v3 = c3[r] * s;
        size_t base = (size_t)m * 64;
        g[base +  0 + l16] = v0;  accum[base +  0 + l16] = v0;
        g[base + 16 + l16] = v1;  accum[base + 16 + l16] = v1;
        g[base + 32 + l16] = v2;  accum[base + 32 + l16] = v2;
        g[base + 48 + l16] = v3;  accum[base + 48 + l16] = v3;
    }
}

// ---------------- edge scatter: accum[dst] += g[src] (L2-resident atomics) ----
__global__ __launch_bounds__(256) void edge_scatter(const int* __restrict__ src,
                                                    const int* __restrict__ dst,
                                                    const float* __restrict__ g,
                                                    float* __restrict__ accum, int nE) {
    int t = blockIdx.x * blockDim.x + threadIdx.x;  // 16 threads per edge (4 floats each)
    int e = t >> 4;
    if (e >= nE) return;
    int q = t & 15;
    int s = src[e], d = dst[e];
    float4 v = ((const float4*)(g + (size_t)s * 64))[q];
    float* out = accum + (size_t)d * 64 + q * 4;
    atomicAdd(out + 0, v.x);
    atomicAdd(out + 1, v.y);
    atomicAdd(out + 2, v.z);
    atomicAdd(out + 3, v.w);
}

// ---------------- epilogue: h = [relu](dinv[i]*accum + b) ----------------
__global__ __launch_bounds__(256) void finish_layer(const float* __restrict__ accum,
                                                    const float* __restrict__ dinv,
                                                    const float* __restrict__ bias,
                                                    float* __restrict__ hout, int n, int do_relu) {
    int t = blockIdx.x * blockDim.x + threadIdx.x;  // one float4 per thread
    if (t >= n * 16) return;
    int i = t >> 4, q = t & 15;
    float s = dinv[i];
    float4 a = ((const float4*)accum)[t];
    float4 b = ((const float4*)bias)[q];
    float4 v;
    v.x = fmaf(a.x, s, b.x);
    v.y = fmaf(a.y, s, b.y);
    v.z = fmaf(a.z, s, b.z);
    v.w = fmaf(a.w, s, b.w);
    if (do_relu) {
        v.x = fmaxf(v.x, 0.0f); v.y = fmaxf(v.y, 0.0f);
        v.z = fmaxf(v.z, 0.0f); v.w = fmaxf(v.w, 0.0f);
    }
    ((float4*)hout)[t] = v;
}

// ---------------- final FC on 32 gathered rows ----------------
__global__ void fc_kernel(const float* __restrict__ h, const int* __restrict__ gidx,
                          const float* __restrict__ W, const float* __restrict__ b,
                          float* __restrict__ out) {
    int gi = blockIdx.x;
    int c = threadIdx.x;
    if (c >= NUM_CLASSES) return;
    const float* hr = h + (size_t)gidx[gi] * HIDDEN;
    float acc = b[c];
    #pragma unroll
    for (int f = 0; f < HIDDEN; f++) acc = fmaf(hr[f], W[f * NUM_CLASSES + c], acc);
    out[gi * NUM_CLASSES + c] = acc;
}

extern "C" void kernel_launch(void* const* d_in, const int* in_sizes, int n_in,
                              void* d_out, int out_size, void* d_ws, size_t ws_size,
                              hipStream_t stream) {
    const float* x     = (const float*)d_in[0];
    const int*   edge  = (const int*)d_in[1];   // [2, nE] flat: src then dst
    const int*   batch = (const int*)d_in[2];
    const int*   mni   = (const int*)d_in[3];
    const float* W1 = (const float*)d_in[4];
    const float* b1 = (const float*)d_in[5];
    const float* W2 = (const float*)d_in[6];
    const float* b2 = (const float*)d_in[7];
    const float* W3 = (const float*)d_in[8];
    const float* b3 = (const float*)d_in[9];
    const float* fcW = (const float*)d_in[10];
    const float* fcb = (const float*)d_in[11];

    const int n  = N_NODES;
    const int nE = N_EDGES;
    const int* src = edge;
    const int* dst = edge + nE;

    // workspace layout (25.6MB x3 feature buffers fit in L2 alongside each other)
    char* ws = (char*)d_ws;
    const size_t featB = (size_t)n * HIDDEN * sizeof(float);        // 25,600,000 B
    float* hA    = (float*)(ws);
    float* g     = (float*)(ws + featB);
    float* accum = (float*)(ws + 2 * featB);
    float* dinv  = (float*)(ws + 3 * featB);
    int*   counts = (int*)(ws + 3 * featB + (((size_t)n * 4 + 255) / 256) * 256);
    int*   gidx   = counts + 64;

    const int B = 256;
    const int nBlk  = (n + B - 1) / B;
    const int eBlk  = (nE + B - 1) / B;
    const int sBlk  = (nE * 16 + B - 1) / B;       // 62500
    const int fBlk  = (n * 16 + B - 1) / B;
    const int gemmBlk = (n / 16 + 7) / 8;          // 8 waves/block, 16 rows/wave

    // norm + batch indexing
    init_deg<<<nBlk, B, 0, stream>>>(dinv, n);
    zero_counts<<<1, 32, 0, stream>>>(counts);
    edge_deg<<<eBlk, B, 0, stream>>>(dst, dinv, nE);
    make_dinv<<<nBlk, B, 0, stream>>>(dinv, n);
    batch_hist<<<nBlk, B, 0, stream>>>(batch, counts, n);
    compute_gidx<<<1, 32, 0, stream>>>(counts, batch, mni, gidx, n);

    // layer 1 (K = NUM_CLASSES)
    gcn_gemm<NUM_CLASSES><<<gemmBlk, 256, 0, stream>>>(x, W1, dinv, g, accum, n);
    edge_scatter<<<sBlk, B, 0, stream>>>(src, dst, g, accum, nE);
    finish_layer<<<fBlk, B, 0, stream>>>(accum, dinv, b1, hA, n, 1);

    // layer 2 (K = HIDDEN)
    gcn_gemm<HIDDEN><<<gemmBlk, 256, 0, stream>>>(hA, W2, dinv, g, accum, n);
    edge_scatter<<<sBlk, B, 0, stream>>>(src, dst, g, accum, nE);
    finish_layer<<<fBlk, B, 0, stream>>>(accum, dinv, b2, hA, n, 1);

    // layer 3 (no relu)
    gcn_gemm<HIDDEN><<<gemmBlk, 256, 0, stream>>>(hA, W3, dinv, g, accum, n);
    edge_scatter<<<sBlk, B, 0, stream>>>(src, dst, g, accum, nE);
    finish_layer<<<fBlk, B, 0, stream>>>(accum, dinv, b3, hA, n, 0);

    // readout
    fc_kernel<<<NUM_GRAPHS, 128, 0, stream>>>(hA, gidx, fcW, fcb, (float*)d_out);
}